// RFFT2d_10728828305643
// MI455X (gfx1250) — compile-verified
//
#include <hip/hip_runtime.h>

typedef __attribute__((ext_vector_type(2))) float v2f;
typedef __attribute__((ext_vector_type(8))) float v8f;

// rFFT-8 over 8x8 blocks of a (32,1,1024,1024) f32 image, expressed as
// D = W(16x8) * X(8x16) via two chained V_WMMA_F32_16X16X4_F32 (K=4 each).
// One wave handles two complete 8x8 blocks (16 segments of 8 floats).
__global__ __launch_bounds__(256) void rfft8_wmma(const float* __restrict__ x,
                                                  float* __restrict__ out) {
  const int      lane = (int)(threadIdx.x & 31u);
  const unsigned wave = (blockIdx.x << 3) | (threadIdx.x >> 5); // tile id, 262144 total

  // tile -> (image, block-row bi, pair-of-blocks bjpair)
  const unsigned bjpair = wave & 63u;        // 64 block-pairs per 1024-wide row
  const unsigned big    = wave >> 6;         // img*128 + bi
  const unsigned img    = big >> 7;          // 0..31
  const unsigned bi     = big & 127u;        // 0..127

  const int n    = lane & 15;                // B/D column (which segment)
  const int half = lane >> 4;                // K-half: lanes 0-15 K=0,1 ; 16-31 K=2,3

  // ---- B operands: column n = rows r=0..7 of block (n>>3) of the pair ----
  const int      blk = n >> 3;               // 0 or 1 within the pair
  const int      r   = n & 7;                // row within the 8x8 block
  const unsigned col = (bjpair * 2u + (unsigned)blk) * 8u;
  const float* sp = x + (((size_t)img << 20) +
                         (size_t)((bi << 3) + (unsigned)r) * 1024u + col);
  const v2f b1 = *(const v2f*)(sp + 2 * half);       // elements {2h, 2h+1}
  const v2f b2 = *(const v2f*)(sp + 2 * half + 4);   // elements {2h+4, 2h+5}

  // ---- A operands: DFT-8 matrix rows m = 2f+c (c=0 real: cos, c=1 imag: -sin),
  //      angle = -2*pi*f*j/8 ; only (f*j) mod 8 matters; scale 1/64 folded in.
  const int f  = n >> 1;
  const int im = n & 1;
  v2f a1, a2;
  {
    const float w = -0.78539816339744831f;   // -pi/4
    const float s = 1.0f / 64.0f;
    const int j0 = 2 * half, j1 = 2 * half + 1;
    const float t0 = w * (float)((f * j0)       & 7);
    const float t1 = w * (float)((f * j1)       & 7);
    const float t2 = w * (float)((f * (j0 + 4)) & 7);
    const float t3 = w * (float)((f * (j1 + 4)) & 7);
    a1.x = (im ? __sinf(t0) : __cosf(t0)) * s;
    a1.y = (im ? __sinf(t1) : __cosf(t1)) * s;
    a2.x = (im ? __sinf(t2) : __cosf(t2)) * s;
    a2.y = (im ? __sinf(t3) : __cosf(t3)) * s;
  }

  // ---- two chained f32 WMMAs: C += A*B over K=0..3 then K=4..7 ----
  v8f c = {};
  c = __builtin_amdgcn_wmma_f32_16x16x4_f32(false, a1, false, b1,
                                            (short)0, c, false, false);
  c = __builtin_amdgcn_wmma_f32_16x16x4_f32(false, a2, false, b2,
                                            (short)0, c, false, false);

  // ---- store: lane n holds m=0..7 of column n; lane n+16 holds m=8,9 ----
  const size_t blkg = (size_t)img * 16384u + (size_t)bi * 128u +
                      (size_t)(bjpair * 2u + (unsigned)blk);
  float* op = out + (blkg * 8u + (unsigned)r) * 10u;   // 40B granularity -> 8B aligned
  if (half == 0) {
    v2f p0 = {c[0], c[1]}, p1 = {c[2], c[3]}, p2 = {c[4], c[5]}, p3 = {c[6], c[7]};
    *(v2f*)(op + 0) = p0;
    *(v2f*)(op + 2) = p1;
    *(v2f*)(op + 4) = p2;
    *(v2f*)(op + 6) = p3;
  } else {
    v2f pt = {c[0], c[1]};                   // m = 8,9 (last complex bin)
    *(v2f*)(op + 8) = pt;
  }
}

extern "C" void kernel_launch(void* const* d_in, const int* in_sizes, int n_in,
                              void* d_out, int out_size, void* d_ws, size_t ws_size,
                              hipStream_t stream) {
  (void)in_sizes; (void)n_in; (void)out_size; (void)d_ws; (void)ws_size;
  const float* x  = (const float*)d_in[0];
  float*       out = (float*)d_out;
  // 262,144 tiles (2 blocks each), 8 waves per 256-thread block -> 32768 blocks.
  hipLaunchKernelGGL(rfft8_wmma, dim3(32768), dim3(256), 0, stream, x, out);
}